// SparseMultiheadAttention_29678224015867
// MI455X (gfx1250) — compile-verified
//
#include <hip/hip_runtime.h>
#include <math.h>

// ---------------------------------------------------------------------------
// Sparse (1.5-entmax) causal multi-head attention, fp32, for gfx1250.
// B=2, T=2048, D=512, H=8, dk=64.
// Uses V_WMMA_F32_16X16X4_F32 for all GEMMs (native fp32 matrix pipe).
// ---------------------------------------------------------------------------

typedef float v2f __attribute__((ext_vector_type(2)));
typedef float v8f __attribute__((ext_vector_type(8)));

#define B_ 2
#define T_ 2048
#define D_ 512
#define H_ 8
#define DK_ 64
#define TPB 256

__device__ __forceinline__ v8f wmma_f32(v2f a, v2f b, v8f c) {
  // 8 args: (neg_a, A, neg_b, B, c_mod, C, reuse_a, reuse_b)
  return __builtin_amdgcn_wmma_f32_16x16x4_f32(false, a, false, b, (short)0, c,
                                               false, false);
}

// A fragment: 16x4 fp32, row-major source with leading dim ld.
// lanes 0-15: {A[m][k0+0], A[m][k0+1]}   lanes 16-31: {A[m][k0+2], A[m][k0+3]}
__device__ __forceinline__ v2f load_a_f32(const float* __restrict__ src, int ld,
                                          int row0, int k0, int lane) {
  int m = lane & 15;
  int kk = (lane >> 4) << 1;
  const float* p = src + (size_t)(row0 + m) * ld + (k0 + kk);
  v2f a;
  a.x = p[0];
  a.y = p[1];
  return a;
}

// B fragment: 4x16 fp32, row-major source with leading dim ld.
// lanes 0-15: {B[k0+0][n], B[k0+1][n]}   lanes 16-31: {B[k0+2][n], B[k0+3][n]}
__device__ __forceinline__ v2f load_b_f32(const float* __restrict__ src, int ld,
                                          int k0, int col0, int lane) {
  int n = lane & 15;
  int kk = (lane >> 4) << 1;
  const float* p = src + (size_t)(k0 + kk) * ld + (col0 + n);
  v2f b;
  b.x = p[0];
  b.y = p[ld];
  return b;
}

// ---------------------------------------------------------------------------
// Kernel 1: fused QKV projection.  C[4096,512] = X @ W{q,k,v} + b{q,k,v},
// scattered into [B,H,T,dk] layout. One 16x16 tile per wave, A reused 3x.
// ---------------------------------------------------------------------------
__global__ __launch_bounds__(256) void proj_qkv_kernel(
    const float* __restrict__ X, const float* __restrict__ Wq,
    const float* __restrict__ bq, const float* __restrict__ Wk,
    const float* __restrict__ bk, const float* __restrict__ Wv,
    const float* __restrict__ bv, float* __restrict__ Qws,
    float* __restrict__ Kws, float* __restrict__ Vws) {
  const int lane = threadIdx.x & 31;
  const int warp = threadIdx.x >> 5;
  const int tile = blockIdx.x * 8 + warp;  // 8192 tiles: 256 m-tiles x 32 n-tiles
  const int mt = tile >> 5;
  const int nt = tile & 31;
  const int row0 = mt * 16, col0 = nt * 16;

  v8f cq = {};
  v8f ck = {};
  v8f cv = {};
  for (int k0 = 0; k0 < D_; k0 += 4) {
    v2f a = load_a_f32(X, D_, row0, k0, lane);
    v2f fq = load_b_f32(Wq, D_, k0, col0, lane);
    cq = wmma_f32(a, fq, cq);
    v2f fk = load_b_f32(Wk, D_, k0, col0, lane);
    ck = wmma_f32(a, fk, ck);
    v2f fv = load_b_f32(Wv, D_, k0, col0, lane);
    cv = wmma_f32(a, fv, cv);
  }

  const int n = lane & 15;
  const int col = col0 + n;
  const int h = col >> 6, d = col & 63;
  const float biasq = bq[col], biask = bk[col], biasv = bv[col];
  const int mbase = row0 + ((lane >> 4) << 3);
#pragma unroll
  for (int r = 0; r < 8; ++r) {
    int row = mbase + r;           // row in [0, B*T)
    int bb = row >> 11;            // / T_
    int t = row & (T_ - 1);
    size_t idx = (((size_t)(bb * H_ + h) * T_) + t) * DK_ + d;
    Qws[idx] = cq[r] + biasq;
    Kws[idx] = ck[r] + biask;
    Vws[idx] = cv[r] + biasv;
  }
}

// ---------------------------------------------------------------------------
// Kernel 2: V_combined = mean over heads of V.  Vc [B,T,dk]
// ---------------------------------------------------------------------------
__global__ void vmean_kernel(const float* __restrict__ Vws,
                             float* __restrict__ Vc) {
  int i = blockIdx.x * blockDim.x + threadIdx.x;
  if (i >= B_ * T_ * DK_) return;
  int d = i & 63;
  int t = (i >> 6) & (T_ - 1);
  int b = i >> 17;  // / (T_*DK_)
  float s = 0.f;
#pragma unroll
  for (int h = 0; h < H_; ++h)
    s += Vws[(((size_t)(b * H_ + h) * T_) + t) * DK_ + d];
  Vc[i] = s * (1.0f / H_);
}

// ---------------------------------------------------------------------------
// Kernel 3: per (b,t) row: scores -> exact 1.5-entmax per head, head-averaged
// weights row written to Wout [B,T,T].
//
// Exactness-preserving truncation: only the first t+1 scores are live (causal);
// the rest are the constant 0.5*NEG_LARGE = -5e8, which can never enter the
// entmax support. So we sort / prefix-scan only n2 = next_pow2(t+1) elements
// (clamped to >=256 so all threads stay busy), cutting average bitonic work
// by ~40%. Columns > t are written as exact zeros.
// ---------------------------------------------------------------------------
__global__ __launch_bounds__(256) void attn_entmax_kernel(
    const float* __restrict__ Qws, const float* __restrict__ Kws,
    float* __restrict__ Wout) {
  __shared__ float q_s[D_];
  __shared__ float zbuf[T_];
  __shared__ float zsbuf[T_];
  __shared__ float s1[T_];
  __shared__ float s2[T_];
  __shared__ float wacc[T_];
  __shared__ float c1[TPB], c2[TPB];
  __shared__ int support;
  __shared__ float tau_star;

  const int tid = threadIdx.x;
  const int b = blockIdx.x >> 11;  // / T_
  const int t = blockIdx.x & (T_ - 1);

  // live-prefix length rounded up to a power of two, at least TPB
  int n2 = TPB;
  while (n2 < t + 1) n2 <<= 1;
  const int chunk = n2 >> 8;  // elements per thread in the scan (1..8)

  for (int i = tid; i < D_; i += TPB) {
    int h = i >> 6, d = i & 63;
    q_s[i] = Qws[(((size_t)(b * H_ + h) * T_) + t) * DK_ + d];
  }
  for (int i = tid; i <= t; i += TPB) wacc[i] = 0.f;
  __syncthreads();

  for (int h = 0; h < H_; ++h) {
    // ---- score row: z = 0.5 * (q.k / sqrt(dk)) = q.k * 0.0625 ----
    const float4* qh = reinterpret_cast<const float4*>(q_s + h * DK_);
    for (int j = tid; j < n2; j += TPB) {
      float zv;
      if (j > t) {
        zv = -5.0e8f;  // 0.5 * NEG_LARGE
      } else {
        const float4* kr = reinterpret_cast<const float4*>(
            Kws + (((size_t)(b * H_ + h) * T_) + j) * DK_);
        float acc = 0.f;
#pragma unroll
        for (int q4 = 0; q4 < DK_ / 4; ++q4) {
          float4 qv = qh[q4];
          float4 kv = kr[q4];
          acc += qv.x * kv.x + qv.y * kv.y + qv.z * kv.z + qv.w * kv.w;
        }
        zv = acc * 0.0625f;
      }
      zbuf[j] = zv;
      zsbuf[j] = zv;
    }
    if (tid == 0) support = 0;
    __syncthreads();

    // ---- bitonic sort descending on zsbuf[0..n2) ----
    for (int k = 2; k <= n2; k <<= 1) {
      for (int jj = k >> 1; jj > 0; jj >>= 1) {
        for (int i = tid; i < n2; i += TPB) {
          int ixj = i ^ jj;
          if (ixj > i) {
            float va = zsbuf[i], vb = zsbuf[ixj];
            bool up = ((i & k) == 0);
            if (up ? (va < vb) : (va > vb)) {
              zsbuf[i] = vb;
              zsbuf[ixj] = va;
            }
          }
        }
        __syncthreads();
      }
    }

    // ---- chunked inclusive prefix sums of zs and zs^2 over [0, n2) ----
    {
      const int base = tid * chunk;
      float r1 = 0.f, r2 = 0.f;
      for (int i2 = 0; i2 < chunk; ++i2) {
        float v = zsbuf[base + i2];
        r1 += v;
        r2 += v * v;
        s1[base + i2] = r1;
        s2[base + i2] = r2;
      }
      c1[tid] = r1;
      c2[tid] = r2;
      __syncthreads();
      // parallel Hillis-Steele inclusive scan over the 256 chunk totals
#pragma unroll
      for (int off = 1; off < TPB; off <<= 1) {
        float a1 = c1[tid], a2 = c2[tid];
        float p1 = (tid >= off) ? c1[tid - off] : 0.f;
        float p2 = (tid >= off) ? c2[tid - off] : 0.f;
        __syncthreads();
        c1[tid] = a1 + p1;
        c2[tid] = a2 + p2;
        __syncthreads();
      }
      float o1 = (tid > 0) ? c1[tid - 1] : 0.f;  // exclusive chunk offset
      float o2 = (tid > 0) ? c2[tid - 1] : 0.f;
      for (int i2 = 0; i2 < chunk; ++i2) {
        s1[base + i2] += o1;
        s2[base + i2] += o2;
      }
    }
    __syncthreads();

    // ---- tau(k), support size, tau_star ----
    {
      int cnt = 0;
      for (int j = tid; j < n2; j += TPB) {
        float kf = (float)(j + 1);
        float mean = s1[j] / kf;
        float msq = s2[j] / kf;
        float ss = kf * (msq - mean * mean);
        float delta = (1.f - ss) / kf;
        float dd = delta > 0.f ? delta : 0.f;
        float tau = mean - sqrtf(dd);
        s1[j] = tau;  // each j owned by exactly one thread
        if (tau <= zsbuf[j]) cnt++;
      }
      atomicAdd(&support, cnt);
    }
    __syncthreads();
    if (tid == 0) tau_star = s1[support - 1];
    __syncthreads();

    // ---- weights = max(z - tau*, 0)^2, accumulate over heads (j <= t) ----
    float ts = tau_star;
    for (int j = tid; j <= t; j += TPB) {
      float w = zbuf[j] - ts;
      w = w > 0.f ? w : 0.f;
      wacc[j] += w * w;
    }
    __syncthreads();
  }

  float* wout = Wout + ((size_t)(b * T_ + t)) * T_;
  for (int j = tid; j < T_; j += TPB)
    wout[j] = (j <= t) ? wacc[j] * (1.0f / H_) : 0.f;
}

// ---------------------------------------------------------------------------
// Kernel 4: AV[b] = weights_avg[b] @ Vc[b]   (2048x2048 @ 2048x64), WMMA fp32
// ---------------------------------------------------------------------------
__global__ __launch_bounds__(256) void av_gemm_kernel(
    const float* __restrict__ Wavg, const float* __restrict__ Vc,
    float* __restrict__ AV) {
  const int lane = threadIdx.x & 31;
  const int warp = threadIdx.x >> 5;
  const int tile = blockIdx.x * 8 + warp;  // 1024 tiles: b(2) x mt(128) x nt(4)
  const int nt = tile & 3;
  const int mt = (tile >> 2) & 127;
  const int b = tile >> 9;
  const float* A = Wavg + (size_t)b * T_ * T_;
  const float* Bm = Vc + (size_t)b * T_ * DK_;
  const int row0 = mt * 16, col0 = nt * 16;

  v8f c = {};
  // causal: A row r of this tile has zeros past column row0+15, so K can stop
  // at the tile's last live column (rounded up to a multiple of 4).
  const int kend = (row0 + 16 + 3) & ~3;
  for (int k0 = 0; k0 < kend; k0 += 4) {
    v2f a = load_a_f32(A, T_, row0, k0, lane);
    v2f f = load_b_f32(Bm, DK_, k0, col0, lane);
    c = wmma_f32(a, f, c);
  }
  const int n = lane & 15;
  const int mb = row0 + ((lane >> 4) << 3);
  float* out = AV + (size_t)b * T_ * DK_;
#pragma unroll
  for (int r = 0; r < 8; ++r)
    out[(size_t)(mb + r) * DK_ + col0 + n] = c[r];
}

// ---------------------------------------------------------------------------
// Kernel 5: Y = AV @ Wo + bo   (4096x64 @ 64x512), WMMA fp32
// ---------------------------------------------------------------------------
__global__ __launch_bounds__(256) void out_gemm_kernel(
    const float* __restrict__ AV, const float* __restrict__ Wo,
    const float* __restrict__ bo, float* __restrict__ Y) {
  const int lane = threadIdx.x & 31;
  const int warp = threadIdx.x >> 5;
  const int tile = blockIdx.x * 8 + warp;  // 8192 tiles: 256 x 32
  const int mt = tile >> 5;
  const int nt = tile & 31;
  const int row0 = mt * 16, col0 = nt * 16;

  v8f c = {};
  for (int k0 = 0; k0 < DK_; k0 += 4) {
    v2f a = load_a_f32(AV, DK_, row0, k0, lane);
    v2f f = load_b_f32(Wo, D_, k0, col0, lane);
    c = wmma_f32(a, f, c);
  }
  const int n = lane & 15;
  const int col = col0 + n;
  const float bias = bo[col];
  const int mb = row0 + ((lane >> 4) << 3);
#pragma unroll
  for (int r = 0; r < 8; ++r)
    Y[(size_t)(mb + r) * D_ + col] = c[r] + bias;
}

// ---------------------------------------------------------------------------
extern "C" void kernel_launch(void* const* d_in, const int* in_sizes, int n_in,
                              void* d_out, int out_size, void* d_ws,
                              size_t ws_size, hipStream_t stream) {
  const float* x = (const float*)d_in[0];
  const float* Wq = (const float*)d_in[1];
  const float* bq = (const float*)d_in[2];
  const float* Wk = (const float*)d_in[3];
  const float* bk = (const float*)d_in[4];
  const float* Wv = (const float*)d_in[5];
  const float* bv = (const float*)d_in[6];
  const float* Wo = (const float*)d_in[7];
  const float* bo = (const float*)d_in[8];

  float* out = (float*)d_out;
  float* Y = out;                                   // [B,T,D]   2,097,152 f
  float* Wavg = out + (size_t)B_ * T_ * D_;         // [B,T,T]   8,388,608 f

  float* ws = (float*)d_ws;
  const size_t qkv = (size_t)B_ * H_ * T_ * DK_;    // 2,097,152 f each
  float* Qws = ws;
  float* Kws = Qws + qkv;
  float* Vws = Kws + qkv;
  float* Vc = Vws + qkv;                            // [B,T,dk]  262,144 f
  float* AV = Vc + (size_t)B_ * T_ * DK_;           // [B,T,dk]  262,144 f

  // 1) QKV projections (WMMA fp32): 8192 tiles / 8 waves per block
  proj_qkv_kernel<<<1024, 256, 0, stream>>>(x, Wq, bq, Wk, bk, Wv, bv, Qws, Kws,
                                            Vws);
  // 2) head-mean of V
  vmean_kernel<<<(B_ * T_ * DK_ + 255) / 256, 256, 0, stream>>>(Vws, Vc);
  // 3) per-row scores + exact 1.5-entmax + head average
  attn_entmax_kernel<<<B_ * T_, 256, 0, stream>>>(Qws, Kws, Wavg);
  // 4) weights_avg @ V_combined (WMMA fp32), causal-truncated K loop
  av_gemm_kernel<<<128, 256, 0, stream>>>(Wavg, Vc, AV);
  // 5) final projection (WMMA fp32)
  out_gemm_kernel<<<1024, 256, 0, stream>>>(AV, Wo, bo, Y);
}